// HeadVer2_14035953123647
// MI455X (gfx1250) — compile-verified
//
#include <hip/hip_runtime.h>

typedef __attribute__((ext_vector_type(2))) float v2f;
typedef __attribute__((ext_vector_type(4))) float v4f;
typedef __attribute__((ext_vector_type(8))) float v8f;

#define B_   8
#define T_   4096
#define C_   1024
#define P_   16                 // T-partitions for parallelism
#define RPP  (T_ / P_)          // 256 rows per partition
#define TILES (RPP / 16)        // 16 wmma tiles per partition

// ---------------- Kernel 1: per-partition column sums ----------------
// sums[b*P_ + p][c] = sum over the 256 rows of partition p of x[b,:,c].
// Regular (RT) loads on purpose: x (134 MB) fits in the 192 MB L2, and the
// scan kernel re-reads it — keep it resident.
__global__ __launch_bounds__(256) void colsum_kernel(const float* __restrict__ x,
                                                     float* __restrict__ sums) {
    int gid = blockIdx.x * 256 + threadIdx.x;       // [0, B_*P_*C_/4)
    int c4 = gid % (C_ / 4);                        // float4 column index
    int bp = gid / (C_ / 4);                        // b*P_ + p
    int p  = bp % P_;
    int b  = bp / P_;
    const v4f* xp = (const v4f*)(x + ((size_t)(b * T_ + p * RPP)) * C_) + c4;
    v4f acc = {};
    #pragma unroll 8
    for (int t = 0; t < RPP; ++t) {
        v4f v = xp[(size_t)t * (C_ / 4)];
        acc.x += v.x; acc.y += v.y; acc.z += v.z; acc.w += v.w;
    }
    ((v4f*)(sums + (size_t)bp * C_))[c4] = acc;
}

// ---------------- Kernel 2: WMMA tile scan ----------------
// One wave32 per (b, 16-column tile, partition). Within each 16x16 tile:
//   D = L16 @ X_tile   via 4x V_WMMA_F32_16X16X4_F32 (L16 = tril(ones))
// giving inclusive prefix sums down the rows; D row 15 is the tile column
// total used to update the running carry.
__global__ __launch_bounds__(256) void wmma_scan_kernel(const float* __restrict__ x,
                                                        const float* __restrict__ sums,
                                                        float* __restrict__ out) {
    const int lane = threadIdx.x & 31;
    const int wave = threadIdx.x >> 5;
    const int w    = blockIdx.x * 8 + wave;         // global wave id
    const int ct   = w % (C_ / 16);                 // column tile
    const int bp   = w / (C_ / 16);
    const int p    = bp % P_;
    const int b    = bp / P_;
    const int c0   = ct * 16;
    const int t0   = p * RPP;

    const int m    = lane & 15;                     // A:M, B/D:N
    const int half = lane >> 4;

    // carry-in: exclusive prefix over preceding partitions for column c0+m
    float carry = 0.0f;
    for (int q = 0; q < p; ++q)
        carry += sums[((size_t)(b * P_ + q)) * C_ + c0 + m];

    // A = 16x16 lower-triangular ones, as 4 K-blocks of 16x4 f32.
    // ISA layout (32-bit A 16x4): lane m holds K = 2*half + {0,1} in v.{x,y}
    v2f a[4];
    #pragma unroll
    for (int j = 0; j < 4; ++j) {
        int k0 = 4 * j + 2 * half;
        a[j].x = (k0     <= m) ? 1.0f : 0.0f;
        a[j].y = (k0 + 1 <= m) ? 1.0f : 0.0f;
    }

    const size_t base = (size_t)b * T_ * C_ + c0 + m;

    for (int tile = 0; tile < TILES; ++tile) {
        const int tb = t0 + tile * 16;

        v8f acc = {};
        #pragma unroll
        for (int j = 0; j < 4; ++j) {
            int k0 = 4 * j + 2 * half;              // row of X within tile
            v2f bf;
            // Each element of x is read exactly once here, and x stays hot in
            // L2 from kernel 1 — mark non-temporal so these lines retire early.
            bf.x = __builtin_nontemporal_load(&x[base + (size_t)(tb + k0)     * C_]);
            bf.y = __builtin_nontemporal_load(&x[base + (size_t)(tb + k0 + 1) * C_]);
            // (neg_a, A, neg_b, B, c_mod, C, reuse_a, reuse_b)
            acc = __builtin_amdgcn_wmma_f32_16x16x4_f32(false, a[j], false, bf,
                                                        (short)0, acc, false, false);
        }

        // Tile column total = D row M=15 -> VGPR7 of lanes 16..31 (ISA C/D map)
        float colsum = __shfl(acc[7], 16 + m, 32);

        #pragma unroll
        for (int r = 0; r < 8; ++r) {
            int Mrow = r + 8 * half;                // ISA C/D map: M = r + 8*half
            float inv = __builtin_amdgcn_rcpf((float)(tb + Mrow + 1));
            // out is write-once, never re-read: NT stores keep it from
            // evicting the L2-resident x between the two passes.
            __builtin_nontemporal_store((acc[r] + carry) * inv,
                                        &out[base + (size_t)(tb + Mrow) * C_]);
        }
        carry += colsum;
    }
}

extern "C" void kernel_launch(void* const* d_in, const int* in_sizes, int n_in,
                              void* d_out, int out_size, void* d_ws, size_t ws_size,
                              hipStream_t stream) {
    const float* x    = (const float*)d_in[0];
    float*       out  = (float*)d_out;
    float*       sums = (float*)d_ws;               // B_*P_*C_ floats = 512 KB

    colsum_kernel<<<(B_ * P_ * (C_ / 4)) / 256, 256, 0, stream>>>(x, sums);

    // 8192 waves total, 8 waves (256 threads) per block
    wmma_scan_kernel<<<(B_ * (C_ / 16) * P_) / 8, 256, 0, stream>>>(x, sums, out);
}